// SAGE_57440892616778
// MI455X (gfx1250) — compile-verified
//
#include <hip/hip_runtime.h>
#include <math.h>

// Problem constants (match reference setup_inputs()).
#define N1c  200000
#define N2c  40000
#define N3c  10000
#define INc  128
#define HIDc 256
#define OUTc 64

typedef float v2f __attribute__((ext_vector_type(2)));
typedef float v8f __attribute__((ext_vector_type(8)));

// ---------------------------------------------------------------------------
// Zero-fill (scratch accumulators must be re-zeroed every call).
// ---------------------------------------------------------------------------
__global__ void zero_f32(float* __restrict__ p, int n) {
    int i = blockIdx.x * blockDim.x + threadIdx.x;
    if (i < n) p[i] = 0.0f;
}

// ---------------------------------------------------------------------------
// Edge scatter: agg[dst[e]][f] += x[src[e]][f]; cnt[dst[e]] += 1 (f==0 lane).
// One thread per (edge, feature). D = 1<<dshift features. x / agg are
// L2-resident (102 MB / 20 MB vs 192 MB L2) so the random gather mostly
// hits L2; atomics resolve at L2 as well.
// ---------------------------------------------------------------------------
__global__ void scatter_edges(const float* __restrict__ x,
                              const int*   __restrict__ src,
                              const int*   __restrict__ dst,
                              float*       __restrict__ agg,
                              float*       __restrict__ cnt,
                              long long total, int dshift) {
    long long i = (long long)blockIdx.x * blockDim.x + threadIdx.x;
    if (i >= total) return;
    int e = (int)(i >> dshift);
    int f = (int)(i & ((1 << dshift) - 1));
    int s = src[e];
    int d = dst[e];
    atomicAdd(&agg[((long long)d << dshift) + f], x[((long long)s << dshift) + f]);
    if (f == 0) atomicAdd(&cnt[d], 1.0f);
}

// ---------------------------------------------------------------------------
// Fused SAGE linear:  out = act( (agg/max(cnt,1)) @ Wl^T + xt @ Wr^T + bias )
//
// One wave per 16x16 output tile, V_WMMA_F32_16X16X4_F32, K stepped by 4.
// Fragment layouts per CDNA5 ISA 7.12.2 (wave32):
//   A 16x4 f32 : lane = (K>>1)*16 + M ; v2f = {A[M][2kh], A[M][2kh+1]}
//   B 4x16 f32 : lane = (K>>1)*16 + N ; v2f = {B[2kh][N], B[2kh+1][N]}
//   C/D 16x16  : VGPR r -> row M = r + 8*(lane>=16), col N = lane&15
// Both GEMM terms accumulate into the same accumulator chain; mean-division,
// bias and ReLU are fused. Grid is exactly tile-divisible -> EXEC all ones.
// ---------------------------------------------------------------------------
__launch_bounds__(128)
__global__ void sage_gemm_wmma(const float* __restrict__ agg,
                               const float* __restrict__ cnt,
                               const float* __restrict__ xt,
                               const float* __restrict__ Wl,
                               const float* __restrict__ Wr,
                               const float* __restrict__ bias,
                               float*       __restrict__ out,
                               int K, int Nout, int relu) {
    const int lane  = threadIdx.x & 31;
    const int wave  = threadIdx.x >> 5;
    const int mTile = blockIdx.x;
    const int nTile = blockIdx.y * 4 + wave;

    const int n  = lane & 15;   // A-row / B-col / C-col index within tile
    const int kh = lane >> 4;   // K-half selector (0 or 1)

    const long long row = (long long)mTile * 16 + n;   // output row for A frag
    const long long col = (long long)nTile * 16 + n;   // output col / W row

    const float inv = 1.0f / fmaxf(cnt[row], 1.0f);

    v8f acc;
    #pragma unroll
    for (int r = 0; r < 8; ++r) acc[r] = 0.0f;

    // Term 1: mean @ Wl^T   (mean fused as agg * inv)
    {
        const float* aRow = agg + row * K;
        const float* bRow = Wl  + col * K;
        for (int k0 = 0; k0 < K; k0 += 4) {
            const int ka = k0 + 2 * kh;
            v2f a; a[0] = aRow[ka] * inv; a[1] = aRow[ka + 1] * inv;
            v2f b; b[0] = bRow[ka];       b[1] = bRow[ka + 1];
            acc = __builtin_amdgcn_wmma_f32_16x16x4_f32(
                false, a, false, b, (short)0, acc, false, false);
        }
    }
    // Term 2: x_tgt @ Wr^T
    {
        const float* aRow = xt + row * K;
        const float* bRow = Wr + col * K;
        for (int k0 = 0; k0 < K; k0 += 4) {
            const int ka = k0 + 2 * kh;
            v2f a; a[0] = aRow[ka]; a[1] = aRow[ka + 1];
            v2f b; b[0] = bRow[ka]; b[1] = bRow[ka + 1];
            acc = __builtin_amdgcn_wmma_f32_16x16x4_f32(
                false, a, false, b, (short)0, acc, false, false);
        }
    }

    const float bv = bias[col];
    const long long mbase = (long long)mTile * 16 + ((lane < 16) ? 0 : 8);
    #pragma unroll
    for (int r = 0; r < 8; ++r) {
        float v = acc[r] + bv;
        if (relu) v = fmaxf(v, 0.0f);
        out[(mbase + r) * Nout + col] = v;
    }
}

// ---------------------------------------------------------------------------
// In-place log_softmax over rows of 64. One wave per row, 2 elems per lane,
// wave32 shuffle reductions.
// ---------------------------------------------------------------------------
__global__ void log_softmax64(float* __restrict__ out, int M) {
    const int lane = threadIdx.x & 31;
    const int wave = threadIdx.x >> 5;
    const int row  = blockIdx.x * 8 + wave;
    if (row >= M) return;
    float* p = out + (long long)row * 64;
    float a = p[lane];
    float b = p[lane + 32];
    float m = fmaxf(a, b);
    #pragma unroll
    for (int off = 16; off >= 1; off >>= 1)
        m = fmaxf(m, __shfl_xor(m, off, 32));
    float s = __expf(a - m) + __expf(b - m);
    #pragma unroll
    for (int off = 16; off >= 1; off >>= 1)
        s += __shfl_xor(s, off, 32);
    const float l = __logf(s) + m;
    p[lane]      = a - l;
    p[lane + 32] = b - l;
}

// ---------------------------------------------------------------------------
// Launch.  Inputs (setup_inputs order):
//  0:x 1:src1 2:dst1 3:src2 4:dst2 5:n2 6:n3 7:W1l 8:W1r 9:b1 10:W2l 11:W2r 12:b2
// ---------------------------------------------------------------------------
extern "C" void kernel_launch(void* const* d_in, const int* in_sizes, int n_in,
                              void* d_out, int out_size, void* d_ws, size_t ws_size,
                              hipStream_t stream) {
    const float* x    = (const float*)d_in[0];
    const int*   src1 = (const int*)  d_in[1];
    const int*   dst1 = (const int*)  d_in[2];
    const int*   src2 = (const int*)  d_in[3];
    const int*   dst2 = (const int*)  d_in[4];
    const float* W1l  = (const float*)d_in[7];
    const float* W1r  = (const float*)d_in[8];
    const float* b1   = (const float*)d_in[9];
    const float* W2l  = (const float*)d_in[10];
    const float* W2r  = (const float*)d_in[11];
    const float* b2   = (const float*)d_in[12];

    const int E1 = in_sizes[1];
    const int E2 = in_sizes[3];

    // Workspace layout (floats): agg1 | cnt1 | h | agg2 | cnt2
    float* agg1 = (float*)d_ws;
    float* cnt1 = agg1 + (size_t)N2c * INc;
    float* h    = cnt1 + N2c;
    float* agg2 = h    + (size_t)N2c * HIDc;
    float* cnt2 = agg2 + (size_t)N3c * HIDc;
    float* out  = (float*)d_out;

    // Zero the two accumulator regions (each contiguous: agg+cnt).
    {
        int z1 = N2c * INc + N2c;
        zero_f32<<<(z1 + 255) / 256, 256, 0, stream>>>(agg1, z1);
        int z2 = N3c * HIDc + N3c;
        zero_f32<<<(z2 + 255) / 256, 256, 0, stream>>>(agg2, z2);
    }

    // Layer 1: aggregate + fused SAGE linear + ReLU -> h [N2, HID]
    {
        long long total = (long long)E1 * INc;
        int blocks = (int)((total + 255) / 256);
        scatter_edges<<<blocks, 256, 0, stream>>>(x, src1, dst1, agg1, cnt1,
                                                  total, 7 /* log2(IN) */);
        dim3 grid(N2c / 16, HIDc / 64);   // 2500 x 4, 4 waves/block cover 64 cols
        sage_gemm_wmma<<<grid, 128, 0, stream>>>(agg1, cnt1, x, W1l, W1r, b1,
                                                 h, INc, HIDc, /*relu=*/1);
    }

    // Layer 2: aggregate + fused SAGE linear -> out [N3, OUT]
    {
        long long total = (long long)E2 * HIDc;
        int blocks = (int)((total + 255) / 256);
        scatter_edges<<<blocks, 256, 0, stream>>>(h, src2, dst2, agg2, cnt2,
                                                  total, 8 /* log2(HID) */);
        dim3 grid(N3c / 16, OUTc / 64);   // 625 x 1
        sage_gemm_wmma<<<grid, 128, 0, stream>>>(agg2, cnt2, h, W2l, W2r, b2,
                                                 out, HIDc, OUTc, /*relu=*/0);
    }

    // Row-wise log_softmax over 64 logits, in place on d_out.
    log_softmax64<<<(N3c + 7) / 8, 256, 0, stream>>>(out, N3c);
}